// GELU115_70428873720403
// MI455X (gfx1250) — compile-verified
//
#include <hip/hip_runtime.h>
#include <cstdint>

#define D_MODEL 1024
#define K_SEL   256
#define WAVE    32
#define WAVES_PER_BLOCK 8
#define BLOCK   (WAVE * WAVES_PER_BLOCK)
#define TOKS_PER_WAVE 2
#define EPL     32              // elements per lane (D_MODEL / WAVE)
#define CHUNK   8               // float4 chunks per lane (EPL / 4)
#define TRASH   256             // dummy histogram bin for non-participants

typedef __attribute__((ext_vector_type(4))) int v4i;

#if defined(__HIP_DEVICE_COMPILE__) && !__has_builtin(__builtin_amdgcn_global_load_async_to_lds_b128)
#warning "gfx1250 async-to-LDS builtin not available on this toolchain; using VGPR staging fallback"
#endif

__device__ __forceinline__ float gelu_exact(float v) {
    return 0.5f * v * (1.0f + erff(v * 0.70710678118654752440f));
}

__device__ __forceinline__ float softplus_f(float v) {
    return fmaxf(v, 0.0f) + log1pf(expf(-fabsf(v)));
}

// Per-lane async copy of 16 bytes global -> LDS (GLOBAL_LOAD_ASYNC_TO_LDS_B128).
__device__ __forceinline__ void async_row16(const float* g, float* l) {
#if __has_builtin(__builtin_amdgcn_global_load_async_to_lds_b128)
    __builtin_amdgcn_global_load_async_to_lds_b128(
        (__attribute__((address_space(1))) v4i*)(uintptr_t)g,
        (__attribute__((address_space(3))) v4i*)(uintptr_t)l,
        0, 0);
#else
    *reinterpret_cast<float4*>(l) = *reinterpret_cast<const float4*>(g);
#endif
}

__device__ __forceinline__ void wait_async_le(int n) {
#if __has_builtin(__builtin_amdgcn_s_wait_asynccnt)
    if (n == 0) __builtin_amdgcn_s_wait_asynccnt(0);
    else        __builtin_amdgcn_s_wait_asynccnt(8);   // allow one prefetched row (8 ops) in flight
#else
    if (n == 0) asm volatile("s_wait_asynccnt 0" ::: "memory");
    else        asm volatile("s_wait_asynccnt 8" ::: "memory");
#endif
}

__global__ __launch_bounds__(BLOCK)
void gelu_rarity_gate_kernel(const float* __restrict__ x,
                             const float* __restrict__ p_log_sigma,
                             const float* __restrict__ p_log_w,
                             const float* __restrict__ ema_prob,
                             float* __restrict__ out,
                             int n_tokens)
{
    // Per-wave private LDS: no cross-wave sharing -> zero __syncthreads in this kernel.
    __shared__ __align__(16) float    xbuf[WAVES_PER_BLOCK][2][D_MODEL]; // 64 KB
    __shared__ __align__(16) unsigned hist[WAVES_PER_BLOCK][272];        // 256 bins + trash, padded

    const int tid  = threadIdx.x;
    const int lane = tid & (WAVE - 1);
    const int wv   = tid / WAVE;

    const float sigma = softplus_f(p_log_sigma[0]) + 0.01f;
    const float w     = softplus_f(p_log_w[0]);

    // rarity = 1 - ema_prob at this lane's 32 feature slots (token-invariant)
    float rar[EPL];
#pragma unroll
    for (int j = 0; j < CHUNK; ++j) {
        const float4 ep = *reinterpret_cast<const float4*>(ema_prob + lane * 4 + 128 * j);
        rar[4 * j + 0] = 1.0f - ep.x;
        rar[4 * j + 1] = 1.0f - ep.y;
        rar[4 * j + 2] = 1.0f - ep.z;
        rar[4 * j + 3] = 1.0f - ep.w;
    }

    const int gw   = blockIdx.x * WAVES_PER_BLOCK + wv;  // global wave id
    const int tok0 = gw * TOKS_PER_WAVE;

    if (tok0 < n_tokens) {
#pragma unroll
        for (int j = 0; j < CHUNK; ++j)
            async_row16(x + (size_t)tok0 * D_MODEL + lane * 4 + 128 * j,
                        &xbuf[wv][0][lane * 4 + 128 * j]);
    }

    for (int i = 0; i < TOKS_PER_WAVE; ++i) {
        const int tok = tok0 + i;
        if (tok >= n_tokens) break;                      // wave-uniform
        const bool more = (i + 1 < TOKS_PER_WAVE) && (tok + 1 < n_tokens);
        if (more) {
#pragma unroll
            for (int j = 0; j < CHUNK; ++j)
                async_row16(x + (size_t)(tok + 1) * D_MODEL + lane * 4 + 128 * j,
                            &xbuf[wv][(i + 1) & 1][lane * 4 + 128 * j]);
            wait_async_le(8);   // async loads complete in order -> oldest row done
        } else {
            wait_async_le(0);
        }

        // Pull this wave's row into registers (8 x ds_load_b128 per lane) and derive keys once.
        float    xv[EPL];
        unsigned key[EPL];
#pragma unroll
        for (int j = 0; j < CHUNK; ++j) {
            const float4 v = *reinterpret_cast<const float4*>(&xbuf[wv][i & 1][lane * 4 + 128 * j]);
            xv[4 * j + 0] = v.x; xv[4 * j + 1] = v.y; xv[4 * j + 2] = v.z; xv[4 * j + 3] = v.w;
        }
#pragma unroll
        for (int e = 0; e < EPL; ++e)
            key[e] = __float_as_uint(xv[e]) & 0x7fffffffu;  // |x| bits: uint order == float order

        // ---- 4-pass radix select of the K-th largest key; wave-local, barrier-free ----
        unsigned need = K_SEL, prefix = 0u, pmask = 0u, tie_total = 0u;
        const int shifts[4] = {23, 15, 7, 0};
#pragma unroll
        for (int p = 0; p < 4; ++p) {
            const int sh = shifts[p];
            // clear own 8 bins (DS ops from one wave are in-order; no barrier needed)
            *reinterpret_cast<uint4*>(&hist[wv][lane * 8])     = make_uint4(0u, 0u, 0u, 0u);
            *reinterpret_cast<uint4*>(&hist[wv][lane * 8 + 4]) = make_uint4(0u, 0u, 0u, 0u);
#pragma unroll
            for (int e = 0; e < EPL; ++e) {
                const unsigned bin = (key[e] >> sh) & 0xffu;
                // branchless: non-participants increment the trash bin (one v_cndmask, no exec juggling)
                const unsigned idx = (p == 0) ? bin
                                   : ((((key[e] ^ prefix) & pmask) == 0u) ? bin : (unsigned)TRASH);
                atomicAdd(&hist[wv][idx], 1u);               // ds_add_u32
            }
            // each lane owns bins [8*lane, 8*lane+7]
            const uint4 h0 = *reinterpret_cast<uint4*>(&hist[wv][lane * 8]);
            const uint4 h1 = *reinterpret_cast<uint4*>(&hist[wv][lane * 8 + 4]);
            const unsigned b[8] = {h0.x, h0.y, h0.z, h0.w, h1.x, h1.y, h1.z, h1.w};
            unsigned tot = 0u;
#pragma unroll
            for (int j = 0; j < 8; ++j) tot += b[j];
            // inclusive suffix-scan toward higher lanes (wave shuffles, no LDS)
            unsigned incl = tot;
#pragma unroll
            for (int off = 1; off < WAVE; off <<= 1) {
                const unsigned v = (unsigned)__shfl_down((int)incl, off, WAVE);
                incl += (lane + off < WAVE) ? v : 0u;
            }
            const unsigned above = incl - tot;           // count in strictly-higher lanes
            // boundary bin within own 8 bins, scanning from the top
            unsigned running = above;
            int      sel_l   = -1;
            unsigned need_l  = 0u, cnt_l = 0u;
#pragma unroll
            for (int j = 7; j >= 0; --j) {
                const unsigned c = running + b[j];
                if (c >= need && running < need) {
                    sel_l = lane * 8 + j; need_l = need - running; cnt_l = b[j];
                }
                running = c;
            }
            const unsigned long long m = __ballot(sel_l >= 0);   // exactly one lane wins
            const int src = (int)(__ffsll((long long)m) - 1);
            const int selbin = __shfl(sel_l, src, WAVE);
            need      = (unsigned)__shfl((int)need_l, src, WAVE);
            tie_total = (unsigned)__shfl((int)cnt_l, src, WAVE); // boundary-bin population
            prefix |= ((unsigned)selbin) << sh;
            pmask  |= 0xffu << sh;
        }
        const unsigned T = prefix;   // exact K-th largest key; tie_total = #elements with key == T

        float lsum = 0.0f;
        if (need == tie_total) {
            // Common case (no straddling ties): take everything >= T. Branchless masked sum.
#pragma unroll
            for (int e = 0; e < EPL; ++e)
                lsum += (key[e] >= T) ? rar[e] : 0.0f;
        } else {
            // Measure-zero case: rank ties deterministically via wave prefix scan.
            unsigned tiecnt = 0u;
#pragma unroll
            for (int e = 0; e < EPL; ++e) tiecnt += (key[e] == T) ? 1u : 0u;
            unsigned tinc = tiecnt;
#pragma unroll
            for (int off = 1; off < WAVE; off <<= 1) {
                const unsigned v = (unsigned)__shfl_up((int)tinc, off, WAVE);
                tinc += (lane >= off) ? v : 0u;
            }
            const unsigned tie_before = tinc - tiecnt;
            unsigned seen = 0u;
#pragma unroll
            for (int e = 0; e < EPL; ++e) {
                if (key[e] > T) {
                    lsum += rar[e];
                } else if (key[e] == T) {
                    if (tie_before + seen < need) lsum += rar[e];
                    ++seen;
                }
            }
        }
        // wave-wide sum (butterfly)
        float tsum = lsum;
#pragma unroll
        for (int off = 16; off >= 1; off >>= 1)
            tsum += __shfl_xor(tsum, off, WAVE);

        const float raw  = tsum * (1.0f / (float)K_SEL);
        const float gate = 1.0f + w * tanhf(sigma * raw);

#pragma unroll
        for (int j = 0; j < CHUNK; ++j) {
            float4 o;
            o.x = gelu_exact(xv[4 * j + 0]) * gate;
            o.y = gelu_exact(xv[4 * j + 1]) * gate;
            o.z = gelu_exact(xv[4 * j + 2]) * gate;
            o.w = gelu_exact(xv[4 * j + 3]) * gate;
            *reinterpret_cast<float4*>(out + (size_t)tok * D_MODEL + lane * 4 + 128 * j) = o;
        }

        // our LDS reads of buf[i&1] must retire before the next prefetch overwrites it
        asm volatile("s_wait_dscnt 0" ::: "memory");
    }
}

extern "C" void kernel_launch(void* const* d_in, const int* in_sizes, int n_in,
                              void* d_out, int out_size, void* d_ws, size_t ws_size,
                              hipStream_t stream) {
    (void)n_in; (void)out_size; (void)d_ws; (void)ws_size;
    const float* x    = (const float*)d_in[0];
    const float* lsig = (const float*)d_in[1];
    const float* lw   = (const float*)d_in[2];
    const float* ema  = (const float*)d_in[3];
    float* out        = (float*)d_out;

    const int n_tokens = in_sizes[0] / D_MODEL;                    // 32768 for reference shapes
    const int toks_per_block = WAVES_PER_BLOCK * TOKS_PER_WAVE;    // 16
    const int grid = (n_tokens + toks_per_block - 1) / toks_per_block;
    gelu_rarity_gate_kernel<<<grid, BLOCK, 0, stream>>>(x, lsig, lw, ema, out, n_tokens);
}